// AttentionLayer_34187939676687
// MI455X (gfx1250) — compile-verified
//
#include <hip/hip_runtime.h>
#include <hip/hip_bf16.h>

typedef __attribute__((ext_vector_type(2))) float v2f;
typedef __attribute__((ext_vector_type(8))) float v8f;

#define N_NODES 8192
#define F_IN    256
#define F_OUT   128
#define DEG     32
#define ALPHA   0.2f
#define NEG_INF (-9.0e15f)

// ---------------------------------------------------------------------------
// Kernel 1: hW = h @ W   (8192x256 @ 256x128, fp32, V_WMMA_F32_16X16X4_F32)
// One wave computes one 16x16 tile. 8 waves per block cover the full N=128.
// grid = 512 blocks (M tiles), block = 256 threads (8 waves).
// ---------------------------------------------------------------------------
__global__ void gemm_hw_kernel(const float* __restrict__ h,
                               const float* __restrict__ W,
                               float* __restrict__ hW) {
    const int lane  = threadIdx.x & 31;
    const int wave  = threadIdx.x >> 5;      // N tile: 0..7
    const int row0  = blockIdx.x * 16;       // M tile origin
    const int col0  = wave * 16;             // N tile origin
    const int r     = lane & 15;             // row (A) / col (B,D) within tile
    const int koff  = (lane >> 4) << 1;      // 0 for lanes 0-15, 2 for 16-31

    const float* __restrict__ hrow = h + (size_t)(row0 + r) * F_IN;

    v8f acc = {};
    #pragma unroll 4
    for (int k = 0; k < F_IN; k += 4) {
        // A 16x4 fp32: VGPR j holds A[r][koff+j]  (contiguous pair -> b64 load)
        v2f a = *(const v2f*)(hrow + k + koff);
        // B 4x16 fp32: VGPR j holds B[koff+j][col]
        v2f b;
        b.x = W[(size_t)(k + koff + 0) * F_OUT + col0 + r];
        b.y = W[(size_t)(k + koff + 1) * F_OUT + col0 + r];
        acc = __builtin_amdgcn_wmma_f32_16x16x4_f32(
            /*neg_a=*/false, a, /*neg_b=*/false, b,
            /*c_mod=*/(short)0, acc, /*reuse_a=*/false, /*reuse_b=*/false);
    }

    // D layout: VGPR v -> M = v + 8*(lane>=16), N = lane%16
    const int mbase = (lane >> 4) * 8;
    float* __restrict__ dst = hW + (size_t)(row0 + mbase) * F_OUT + col0 + r;
    #pragma unroll
    for (int v = 0; v < 8; ++v)
        dst[(size_t)v * F_OUT] = acc[v];
}

// ---------------------------------------------------------------------------
// Kernel 2: h1[i] = hW[i,:] . a[0:128]   h2[i] = hW[i,:] . a[128:256]
// One wave per row; lane owns 4 features; shfl_xor tree reduction.
// grid = 1024, block = 256 (8 rows per block).
// ---------------------------------------------------------------------------
__global__ void rowdots_kernel(const float* __restrict__ hW,
                               const float* __restrict__ a,
                               float* __restrict__ h1,
                               float* __restrict__ h2) {
    const int lane = threadIdx.x & 31;
    const int row  = blockIdx.x * 8 + (threadIdx.x >> 5);

    const float4 x  = ((const float4*)(hW + (size_t)row * F_OUT))[lane];
    const float4 a1 = ((const float4*)a)[lane];
    const float4 a2 = ((const float4*)(a + F_OUT))[lane];

    float d1 = x.x * a1.x + x.y * a1.y + x.z * a1.z + x.w * a1.w;
    float d2 = x.x * a2.x + x.y * a2.y + x.z * a2.z + x.w * a2.w;
    #pragma unroll
    for (int off = 16; off; off >>= 1) {
        d1 += __shfl_xor(d1, off, 32);
        d2 += __shfl_xor(d2, off, 32);
    }
    if (lane == 0) { h1[row] = d1; h2[row] = d2; }
}

// ---------------------------------------------------------------------------
// Kernel 3: sparse masked softmax attention + aggregation + ELU.
// One wave per row i; lane = edge slot. Duplicate neighbors fold into the
// mask count; self-loop merged per the `+ eye` semantics of the reference.
// grid = 1024, block = 256 (8 rows per block).
// ---------------------------------------------------------------------------
__global__ void aggregate_kernel(const float* __restrict__ hW,
                                 const float* __restrict__ h1,
                                 const float* __restrict__ h2,
                                 const int* __restrict__ nbr,
                                 float* __restrict__ out) {
    const int lane = threadIdx.x & 31;
    const int i    = blockIdx.x * 8 + (threadIdx.x >> 5);

    const int j = nbr[(size_t)i * DEG + lane];

    // count duplicates of j within the wave; keep only the first occurrence
    int  cnt   = 0;
    bool first = true;
    #pragma unroll
    for (int t = 0; t < 32; ++t) {
        const int jt = __shfl(j, t, 32);
        if (jt == j) { ++cnt; if (t < lane) first = false; }
    }
    const bool self_present =
        ((unsigned)(__ballot(j == i) & 0xffffffffull)) != 0u;

    const float h1i = h1[i];

    // mask[i,j] = (cnt/32 + (j==i))/2 ; score = LeakyReLU(mask*(h1[i]+h2[j]))
    float sc = (0.03125f * (float)cnt + (j == i ? 1.0f : 0.0f)) * 0.5f
             * (h1i + h2[j]);
    sc = sc > 0.0f ? sc : ALPHA * sc;

    // extra self-loop entry only if i is not already among the neighbors
    float sself;
    {
        float t = 0.5f * (h1i + h2[i]);
        t = t > 0.0f ? t : ALPHA * t;
        sself = (!self_present && t != 0.0f) ? t : NEG_INF;
    }

    // entries that are exactly 0 are masked out (att = NEG_INF in reference)
    const float myval = (first && sc != 0.0f) ? sc : NEG_INF;

    float m = myval;
    #pragma unroll
    for (int off = 16; off; off >>= 1) m = fmaxf(m, __shfl_xor(m, off, 32));
    m = fmaxf(m, sself);

    const float es    = (myval > 0.5f * NEG_INF) ? __expf(myval - m) : 0.0f;
    const float eself = (sself > 0.5f * NEG_INF) ? __expf(sself - m) : 0.0f;

    float denom = es;
    #pragma unroll
    for (int off = 16; off; off >>= 1) denom += __shfl_xor(denom, off, 32);
    denom += eself;

    const float w     = es / denom;
    const float wself = eself / denom;

    // aggregation: lane owns features [4*lane, 4*lane+4)
    const int f = lane * 4;
    float acc0 = 0.f, acc1 = 0.f, acc2 = 0.f, acc3 = 0.f;
    for (int t = 0; t < 32; ++t) {
        const float wt = __shfl(w, t, 32);
        const int   jt = __shfl(j, t, 32);
        if (wt != 0.0f) {                       // uniform across the wave
            const float4 v = *(const float4*)(hW + (size_t)jt * F_OUT + f);
            acc0 += wt * v.x; acc1 += wt * v.y;
            acc2 += wt * v.z; acc3 += wt * v.w;
        }
    }
    if (wself != 0.0f) {
        const float4 v = *(const float4*)(hW + (size_t)i * F_OUT + f);
        acc0 += wself * v.x; acc1 += wself * v.y;
        acc2 += wself * v.z; acc3 += wself * v.w;
    }

    // ELU
    float4 o;
    o.x = acc0 > 0.0f ? acc0 : __expf(acc0) - 1.0f;
    o.y = acc1 > 0.0f ? acc1 : __expf(acc1) - 1.0f;
    o.z = acc2 > 0.0f ? acc2 : __expf(acc2) - 1.0f;
    o.w = acc3 > 0.0f ? acc3 : __expf(acc3) - 1.0f;
    *(float4*)(out + (size_t)i * F_OUT + f) = o;
}

// ---------------------------------------------------------------------------
extern "C" void kernel_launch(void* const* d_in, const int* in_sizes, int n_in,
                              void* d_out, int out_size, void* d_ws, size_t ws_size,
                              hipStream_t stream) {
    (void)in_sizes; (void)n_in; (void)out_size; (void)ws_size;

    const float* h   = (const float*)d_in[0];   // [8192, 256]
    const float* W   = (const float*)d_in[1];   // [256, 128]
    const float* a   = (const float*)d_in[2];   // [256, 1]
    const int*   nbr = (const int*)  d_in[3];   // [8192, 32]
    float*       out = (float*)d_out;           // [8192, 128]

    float* hW = (float*)d_ws;                       // 8192*128 floats (4 MB)
    float* h1 = hW + (size_t)N_NODES * F_OUT;       // 8192 floats
    float* h2 = h1 + N_NODES;                       // 8192 floats

    gemm_hw_kernel  <<<N_NODES / 16, 256, 0, stream>>>(h, W, hW);
    rowdots_kernel  <<<N_NODES / 8,  256, 0, stream>>>(hW, a, h1, h2);
    aggregate_kernel<<<N_NODES / 8,  256, 0, stream>>>(hW, h1, h2, nbr, out);
}